// GrimmeD3EnergyLayer_22763326669101
// MI455X (gfx1250) — compile-verified
//
#include <hip/hip_runtime.h>
#include <hip/hip_bf16.h>

// ---------------- Grimme D3 constants ----------------
#define D3_A1      0.3385f
#define D3_A2      2.883f
#define D3_S6      1.0f
#define D3_S8      0.9171f
#define D3_K1      16.0f
#define D3_K3      -4.0f
#define BOHR_IN_R  0.5291772108f
#define MAX_ZA     95
#define GRID_PTS   25      // 5x5 CN reference grid
#define THREADS    256     // 8 wave32 waves per block

#if __has_builtin(__builtin_amdgcn_global_load_async_to_lds_b32)
typedef __attribute__((address_space(1))) int g_int;   // global (__device__)
typedef __attribute__((address_space(3))) int l_int;   // LDS (__shared__)
#define HAVE_ASYNC_LDS 1
#else
#define HAVE_ASYNC_LDS 0
#endif

// Stage a tiny (<=95 element) float table into LDS using the CDNA5 async
// global->LDS path when available; fall back to a plain copy otherwise.
#if HAVE_ASYNC_LDS
#define STAGE_TABLE_TO_LDS(dst_lds, src_glb)                                    \
  do {                                                                          \
    if (threadIdx.x < MAX_ZA) {                                                 \
      __builtin_amdgcn_global_load_async_to_lds_b32(                            \
          (g_int*)((src_glb) + threadIdx.x),                                    \
          (l_int*)((dst_lds) + threadIdx.x), 0, 0);                             \
    }                                                                           \
    asm volatile("s_wait_asynccnt 0" ::: "memory");                             \
    __syncthreads();                                                            \
  } while (0)
#else
#define STAGE_TABLE_TO_LDS(dst_lds, src_glb)                                    \
  do {                                                                          \
    if (threadIdx.x < MAX_ZA) (dst_lds)[threadIdx.x] = (src_glb)[threadIdx.x];  \
    __syncthreads();                                                            \
  } while (0)
#endif

// --------------------------------------------------------------------------
// Zero the per-atom accumulators (nc workspace + output energies).
// --------------------------------------------------------------------------
__global__ void d3_zero_kernel(float* __restrict__ nc,
                               float* __restrict__ out,
                               int n_atoms) {
  int i = blockIdx.x * blockDim.x + threadIdx.x;
  if (i < n_atoms) {
    nc[i]  = 0.0f;
    out[i] = 0.0f;
  }
}

// --------------------------------------------------------------------------
// Pass 1: coordination numbers.
//   damp = 1 / (1 + exp(-K1 * (rco/D - 1)))  scatter-added into nc[idx_i].
// --------------------------------------------------------------------------
__global__ void d3_cn_kernel(const int*   __restrict__ Za,
                             const float* __restrict__ Dij,
                             const int*   __restrict__ idx_i,
                             const int*   __restrict__ idx_j,
                             const float* __restrict__ rcov,
                             float*       __restrict__ nc,
                             int n_pair) {
  __shared__ float s_rcov[MAX_ZA + 1];
  STAGE_TABLE_TO_LDS(s_rcov, rcov);

  int p = blockIdx.x * blockDim.x + threadIdx.x;
  if (p >= n_pair) return;

  int   ai  = idx_i[p];
  int   aj  = idx_j[p];
  int   zi  = Za[ai];
  int   zj  = Za[aj];
  float D   = Dij[p] / BOHR_IN_R;            // distance in bohr
  float rco = s_rcov[zi] + s_rcov[zj];
  float damp = 1.0f / (1.0f + __expf(-D3_K1 * (rco / D - 1.0f)));
  atomicAdd(&nc[ai], damp);
}

// --------------------------------------------------------------------------
// Pass 2: softmax-interpolated C6, BJ-damped -C6/r^6 - C8/r^8, scatter-add.
// Online (streaming) softmax over the 25 reference grid points: each c6ab
// triplet is read exactly once; running max + rescale matches jax.nn.softmax
// (which subtracts the max) in structure.
// --------------------------------------------------------------------------
__global__ void d3_energy_kernel(const int*   __restrict__ Za,
                                 const float* __restrict__ Dij,
                                 const int*   __restrict__ idx_i,
                                 const int*   __restrict__ idx_j,
                                 const float* __restrict__ c6ab,
                                 const float* __restrict__ r2r4,
                                 const float* __restrict__ nc,
                                 float*       __restrict__ out,
                                 int n_pair) {
  __shared__ float s_r2r4[MAX_ZA + 1];
  STAGE_TABLE_TO_LDS(s_r2r4, r2r4);

  int p = blockIdx.x * blockDim.x + threadIdx.x;
  if (p >= n_pair) return;

  int   ai  = idx_i[p];
  int   aj  = idx_j[p];
  int   zi  = Za[ai];
  int   zj  = Za[aj];
  float nci = nc[ai];
  float ncj = nc[aj];

  // 25 x {c6_ref, cn_i, cn_j} block for this (Zi, Zj) species pair: 300 bytes,
  // spanning 3 cachelines. Prefetch all of them ahead of the unrolled loop.
  const float* tab = c6ab + (size_t)(zi * MAX_ZA + zj) * (GRID_PTS * 3);
  __builtin_prefetch(tab,      0, 0);   // -> global_prefetch_b8
  __builtin_prefetch(tab + 32, 0, 0);
  __builtin_prefetch(tab + 64, 0, 0);

  // Online softmax: m = running max logit, ssum = sum exp(l - m),
  // wc6 = sum exp(l - m) * c6_ref.
  float m    = -3.0e38f;
  float ssum = 0.0f;
  float wc6  = 0.0f;
#pragma unroll
  for (int k = 0; k < GRID_PTS; ++k) {
    float c6ref = tab[3 * k + 0];
    float dci   = tab[3 * k + 1] - nci;
    float dcj   = tab[3 * k + 2] - ncj;
    float logit = (c6ref > 0.0f) ? (D3_K3 * (dci * dci + dcj * dcj)) : -1.0e10f;
    float mn = fmaxf(m, logit);
    float sc = __expf(m - mn);       // rescale old accumulators
    float w  = __expf(logit - mn);
    ssum = ssum * sc + w;
    wc6  = wc6  * sc + w * c6ref;
    m = mn;
  }
  float c6 = wc6 / ssum;
  float c8 = 3.0f * c6 * s_r2r4[zi] * s_r2r4[zj];

  float D  = Dij[p] / BOHR_IN_R;
  float r2 = D * D;
  float r6 = r2 * r2 * r2;
  float r8 = r6 * r2;

  float tmp  = D3_A1 * sqrtf(c8 / (c6 + 1e-10f) + 1e-10f) + D3_A2;
  float tmp2 = tmp * tmp;
  float tmp6 = tmp2 * tmp2 * tmp2;
  float tmp8 = tmp6 * tmp2;

  float e6 = -0.5f * D3_S6 * c6 / (r6 + tmp6);
  float e8 = -0.5f * D3_S8 * c8 / (r8 + tmp8);
  atomicAdd(&out[ai], e6 + e8);
}

// --------------------------------------------------------------------------
// Launch: zero accumulators, CN pass, energy pass (stream-ordered => the
// kernel boundary is the required global sync between the two scatter passes).
// --------------------------------------------------------------------------
extern "C" void kernel_launch(void* const* d_in, const int* in_sizes, int n_in,
                              void* d_out, int out_size, void* d_ws, size_t ws_size,
                              hipStream_t stream) {
  const int*   Za    = (const int*)  d_in[0];
  const float* Dij   = (const float*)d_in[1];
  const int*   idx_i = (const int*)  d_in[2];
  const int*   idx_j = (const int*)  d_in[3];
  const float* c6ab  = (const float*)d_in[4];
  const float* rcov  = (const float*)d_in[5];
  const float* r2r4  = (const float*)d_in[6];
  float*       out   = (float*)d_out;

  int n_atoms = in_sizes[0];
  int n_pair  = in_sizes[1];

  float* nc = (float*)d_ws;   // n_atoms floats of scratch

  int blocks_atoms = (n_atoms + THREADS - 1) / THREADS;
  int blocks_pairs = (n_pair  + THREADS - 1) / THREADS;

  d3_zero_kernel<<<blocks_atoms, THREADS, 0, stream>>>(nc, out, n_atoms);
  d3_cn_kernel<<<blocks_pairs, THREADS, 0, stream>>>(Za, Dij, idx_i, idx_j,
                                                     rcov, nc, n_pair);
  d3_energy_kernel<<<blocks_pairs, THREADS, 0, stream>>>(Za, Dij, idx_i, idx_j,
                                                         c6ab, r2r4, nc, out,
                                                         n_pair);
}